// CollisionLoss_80032420594331
// MI455X (gfx1250) — compile-verified
//
#include <hip/hip_runtime.h>
#include <cstdint>

#define MAP_H      200
#define MAP_W      200
#define MAP_ELEMS  (MAP_H * MAP_W)     // 40000 floats = 160 KB, fits CDNA5 320KB LDS
#define BLOCK      512

// Builtin signature (revealed by round-1 diagnostic):
//   void __builtin_amdgcn_global_load_async_to_lds_b128(v4i AS1*, v4i AS3*, imm int, imm int)
typedef int v4i_t __attribute__((vector_size(16)));
typedef __attribute__((address_space(1))) v4i_t* gv4i_p;   // global int4*
typedef __attribute__((address_space(3))) v4i_t* lv4i_p;   // LDS int4*

#if __has_builtin(__builtin_amdgcn_global_load_async_to_lds_b128)
#define USE_ASYNC_BUILTIN 1
#pragma message("CDNA5 path: __builtin_amdgcn_global_load_async_to_lds_b128 available")
#else
#define USE_ASYNC_BUILTIN 0
#pragma message("CDNA5 path: async builtin NOT available -> inline asm global_load_async_to_lds_b128")
#endif

__device__ __forceinline__ uint32_t lds_addr_of(const void* p) {
    // generic -> LDS addrspace cast, then to 32-bit LDS byte offset
    return (uint32_t)(uint64_t)(__attribute__((address_space(3))) const void*)p;
}

__device__ __forceinline__ void wait_asynccnt0() {
#if __has_builtin(__builtin_amdgcn_s_wait_asynccnt)
    __builtin_amdgcn_s_wait_asynccnt(0);
#else
    asm volatile("s_wait_asynccnt 0" ::: "memory");
#endif
}

__global__ __launch_bounds__(BLOCK)
void collision_partial_kernel(const float* __restrict__ opState,
                              const float* __restrict__ envs,
                              float* __restrict__ partials,
                              int N)
{
    __shared__ float s_esdf[MAP_ELEMS];         // 160 KB
    __shared__ float s_warp[BLOCK / 32];

    const int b   = blockIdx.x;
    const int tid = threadIdx.x;

    // ---------- stage this batch's 200x200 ESDF into LDS (async, 16B/lane) ----------
    const float* map = envs + (size_t)b * MAP_ELEMS;
    for (int i = tid * 4; i < MAP_ELEMS; i += BLOCK * 4) {
#if USE_ASYNC_BUILTIN
        __builtin_amdgcn_global_load_async_to_lds_b128(
            (gv4i_p)(map + i), (lv4i_p)(&s_esdf[i]), /*offset=*/0, /*cpol=*/0);
#else
        asm volatile("global_load_async_to_lds_b128 %0, %1, off"
                     :: "v"(lds_addr_of(&s_esdf[i])), "v"(map + i)
                     : "memory");
#endif
    }
    wait_asynccnt0();
    __syncthreads();

    // ---------- stream opState, bilinear-gather from LDS, accumulate penalty ----------
    const float2* op = reinterpret_cast<const float2*>(opState) + (size_t)b * N;
    float acc = 0.0f;
    for (int n = tid; n < N; n += BLOCK) {
        __builtin_prefetch(op + n + BLOCK, 0, 0);   // global_prefetch (speculative)
        const float2 p = op[n];

        const bool oor = (p.x < -9.9f) | (p.x > 9.9f) | (p.y < -9.9f) | (p.y > 9.9f);
        const float px = fminf(fmaxf(p.x, -9.9f), 9.9f);
        const float py = fminf(fmaxf(p.y, -9.9f), 9.9f);

        // idx = floor((pos - 0.5*RES + 10)/RES)
        const int ix = (int)floorf(((px - 0.05f) + 10.0f) / 0.1f);
        const int iy = (int)floorf(((py - 0.05f) + 10.0f) / 0.1f);
        const float cx = ((float)ix + 0.5f) * 0.1f - 10.0f;
        const float cy = ((float)iy + 0.5f) * 0.1f - 10.0f;
        const float dx = (px - cx) / 0.1f;
        const float dy = (py - cy) / 0.1f;

        const int base  = ix * MAP_W + iy;
        const float v00 = s_esdf[base];
        const float v10 = s_esdf[base + MAP_W];
        const float v01 = s_esdf[base + 1];
        const float v11 = s_esdf[base + MAP_W + 1];

        const float vx0 = (1.0f - dx) * v00 + dx * v10;
        const float vx1 = (1.0f - dx) * v01 + dx * v11;
        const float v   = (1.0f - dy) * vx0 + dy * vx1;

        const float dist = oor ? -1.0f : v;
        const float viod = 10.0f * (0.3f - dist);
        const float r    = fmaxf(viod, 0.0f);
        acc = fmaf(r, r, acc);
    }

    // ---------- wave32 + cross-wave reduction, one partial per block ----------
    for (int off = 16; off > 0; off >>= 1) acc += __shfl_down(acc, off, 32);
    const int lane = tid & 31, wid = tid >> 5;
    if (lane == 0) s_warp[wid] = acc;
    __syncthreads();
    if (wid == 0) {
        float v = (lane < (BLOCK / 32)) ? s_warp[lane] : 0.0f;
        for (int off = 8; off > 0; off >>= 1) v += __shfl_down(v, off, 32);
        if (lane == 0) partials[b] = v;
    }
}

__global__ __launch_bounds__(BLOCK)
void collision_finalize_kernel(const float* __restrict__ partials,
                               float* __restrict__ out,
                               int nPart, float invCount)
{
    __shared__ float s_warp[BLOCK / 32];
    const int tid = threadIdx.x;
    float v = (tid < nPart) ? partials[tid] : 0.0f;
    for (int off = 16; off > 0; off >>= 1) v += __shfl_down(v, off, 32);
    const int lane = tid & 31, wid = tid >> 5;
    if (lane == 0) s_warp[wid] = v;
    __syncthreads();
    if (wid == 0) {
        float w = (lane < (BLOCK / 32)) ? s_warp[lane] : 0.0f;
        for (int off = 8; off > 0; off >>= 1) w += __shfl_down(w, off, 32);
        if (lane == 0) out[0] = w * invCount;
    }
}

extern "C" void kernel_launch(void* const* d_in, const int* in_sizes, int n_in,
                              void* d_out, int out_size, void* d_ws, size_t ws_size,
                              hipStream_t stream) {
    const float* opState = (const float*)d_in[0];   // (B, N, 2) f32
    const float* envs    = (const float*)d_in[1];   // (B, 1, 200, 200) f32
    float* out      = (float*)d_out;                // scalar
    float* partials = (float*)d_ws;                 // B floats of scratch

    const int B = in_sizes[1] / MAP_ELEMS;          // 512
    const int N = in_sizes[0] / (2 * B);            // 8192

    collision_partial_kernel<<<dim3(B), dim3(BLOCK), 0, stream>>>(opState, envs, partials, N);
    const float invCount = 1.0f / ((float)B * (float)N);
    collision_finalize_kernel<<<dim3(1), dim3(BLOCK), 0, stream>>>(partials, out, B, invCount);
}